// Star_Net_old_38104949850317
// MI455X (gfx1250) — compile-verified
//
#include <hip/hip_runtime.h>

#define B_  65536
#define D_  8
#define E_  16
#define H_  80
#define F_  9
#define V_  100000
#define IN_ 160   // E*(F+1)
#define KP_ (IN_/2)   // 80 K-pairs

typedef float v2f __attribute__((ext_vector_type(2)));
typedef float v8f __attribute__((ext_vector_type(8)));

// LDS row stride for the 32x160 activation tile: 164 gives conflict-free
// banks for 16 consecutive rows (36*r mod 64 all distinct) and 16B alignment.
#define XS_ 164
#define TM_ 32    // rows per block (two 16-row WMMA tiles)

// ---------------------------------------------------------------------------
// K0: star-merge weights into workspace in WMMA-fragment layout, zero counters.
// w1p[d][p][n] = { W1d[2p][n], W1d[2p+1][n] }  (float2), p = K-pair index.
// ---------------------------------------------------------------------------
__global__ void starnet_prep(const float* __restrict__ sw1, const float* __restrict__ dw1,
                             const float* __restrict__ sb1, const float* __restrict__ db1,
                             const float* __restrict__ sw2, const float* __restrict__ dw2,
                             const float* __restrict__ sb2, const float* __restrict__ db2,
                             float2* __restrict__ w1p, float* __restrict__ b1cat,
                             float* __restrict__ w2cat, float* __restrict__ b2cat,
                             int* __restrict__ cnt) {
  int idx = blockIdx.x * 256 + threadIdx.x;
  if (idx < D_ * KP_ * H_) {
    int d   = idx / (KP_ * H_);
    int rem = idx % (KP_ * H_);
    int p   = rem / H_;
    int n   = rem % H_;
    int i0  = (2 * p) * H_ + n;        // within one (IN,H) matrix
    int i1  = i0 + H_;
    long base = (long)d * (IN_ * H_);
    float2 v;
    v.x = sw1[i0] * dw1[base + i0];
    v.y = sw1[i1] * dw1[base + i1];
    w1p[idx] = v;
  }
  if (idx < D_ * H_) {
    b1cat[idx] = sb1[idx % H_] + db1[idx];
    w2cat[idx] = sw2[idx % H_] * dw2[idx];
  }
  if (idx < D_)  b2cat[idx] = sb2[0] + db2[idx];
  if (idx < 16)  cnt[idx] = 0;
}

// ---------------------------------------------------------------------------
// K1: per-sample domain logit (covers padding samples fully) + bucket samples
// by domain into permutation lists.
// ---------------------------------------------------------------------------
__global__ void starnet_route(const int* __restrict__ pid,
                              const float* __restrict__ emb_pid,
                              const float* __restrict__ dl_w,
                              const float* __restrict__ dl_b,
                              float* __restrict__ out,
                              int* __restrict__ cnt, int* __restrict__ perm) {
  int b = blockIdx.x * 256 + threadIdx.x;
  if (b >= B_) return;
  int p = pid[b];
  const float* row = emb_pid + p * E_;
  float s = dl_b[0];
#pragma unroll
  for (int e = 0; e < E_; ++e) s += fmaxf(row[e], 0.0f) * dl_w[e];
  out[b] = s;                      // logits for pid==0 are exactly this
  if (p > 0) {
    int pos = atomicAdd(&cnt[p - 1], 1);
    perm[(p - 1) * B_ + pos] = b;
  }
}

__device__ __forceinline__ float4 relu4(float4 v) {
  v.x = fmaxf(v.x, 0.0f); v.y = fmaxf(v.y, 0.0f);
  v.z = fmaxf(v.z, 0.0f); v.w = fmaxf(v.w, 0.0f);
  return v;
}

// ---------------------------------------------------------------------------
// K2: per-domain fused gather + (32x160)@(160x80) WMMA GEMM + bias/relu +
// layer-2 dot + scatter-add. One wave32 per 32-sample tile; each B fragment
// (one global_load_b64) feeds two v_wmma_f32_16x16x4_f32.
// ---------------------------------------------------------------------------
__global__ __launch_bounds__(32)
void starnet_gemm(const int* __restrict__ pid, const int* __restrict__ feats,
                  const float* __restrict__ emb_pid, const float* __restrict__ emb_feats,
                  const float2* __restrict__ w1p, const float* __restrict__ b1cat,
                  const float* __restrict__ w2cat, const float* __restrict__ b2cat,
                  const int* __restrict__ cnt, const int* __restrict__ perm,
                  float* __restrict__ out) {
  const int lane = threadIdx.x;
  const int d    = blockIdx.y;
  const int tile = blockIdx.x;
  const int cd   = cnt[d];
  if (tile * TM_ >= cd) return;

  __shared__ int   sidx[TM_];
  __shared__ float Xs[TM_ * XS_];

  {
    int p = tile * TM_ + lane;
    sidx[lane] = (p < cd) ? perm[d * B_ + p] : -1;
  }
  __syncthreads();

  // Gather 320 embedding rows (32 samples x 10 slots), relu, into LDS.
  // Pair index pr = m*10 + j : j==0 -> pid embedding, j>0 -> feature j-1.
#pragma unroll
  for (int it = 0; it < TM_ * 10 / 32; ++it) {
    int pr = lane + 32 * it;
    int m  = pr / 10;
    int j  = pr % 10;
    int b  = sidx[m];
    float4 v0 = {0.f, 0.f, 0.f, 0.f}, v1 = v0, v2 = v0, v3 = v0;
    if (b >= 0) {
      const float* src;
      if (j == 0) {
        src = emb_pid + (long)pid[b] * E_;
      } else {
        long r = (long)(j - 1) * (V_ + 1) + (long)feats[b * F_ + (j - 1)];
        src = emb_feats + r * E_;
      }
      const float4* s4 = (const float4*)src;
      v0 = relu4(s4[0]); v1 = relu4(s4[1]); v2 = relu4(s4[2]); v3 = relu4(s4[3]);
    }
    float4* dst = (float4*)(Xs + m * XS_ + j * 16);
    dst[0] = v0; dst[1] = v1; dst[2] = v2; dst[3] = v3;
  }
  __syncthreads();

  // Accumulators: 2 M-groups x 5 N-tiles of 16x16 f32
  // (C/D layout: lane<16 -> M=vgpr, lane>=16 -> M=vgpr+8, N = lane%16).
  v8f acc[2][5];
#pragma unroll
  for (int g = 0; g < 2; ++g)
#pragma unroll
    for (int t = 0; t < 5; ++t)
#pragma unroll
      for (int e = 0; e < 8; ++e) acc[g][t][e] = 0.0f;

  const int half = lane >> 4;       // 0: K {k,k+1}, 1: K {k+2,k+3}
  const int col  = lane & 15;       // A: row M ; B: column N (within tile)
  const float2* W1d = w1p + (long)d * (KP_ * H_);

  for (int k = 0; k < IN_; k += 4) {
    int kk = k + 2 * half;
    int p  = (k >> 1) + half;       // K-pair index for this lane
    // A fragments (16x4 f32 layout, ISA 7.12.2): {X[M][kk], X[M][kk+1]}
    v2f a0 = *(const v2f*)(Xs + col * XS_ + kk);
    v2f a1 = *(const v2f*)(Xs + (col + 16) * XS_ + kk);
#pragma unroll
    for (int t = 0; t < 5; ++t) {
      // B fragment (4x16): one b64 load {W[kk][n], W[kk+1][n]}, n = t*16+col
      float2 w = W1d[p * H_ + t * 16 + col];
      v2f bb; bb.x = w.x; bb.y = w.y;
      acc[0][t] = __builtin_amdgcn_wmma_f32_16x16x4_f32(
          false, a0, false, bb, (short)0, acc[0][t], false, false);
      acc[1][t] = __builtin_amdgcn_wmma_f32_16x16x4_f32(
          false, a1, false, bb, (short)0, acc[1][t], false, false);
    }
  }

  // Layer 2 fused on the accumulator layout:
  // logit[M] = sum_n relu(h[M][n] + b1[n]) * w2[n]  (+ b2), n = t*16 + col.
  float b1v[5], w2v[5];
#pragma unroll
  for (int t = 0; t < 5; ++t) {
    b1v[t] = b1cat[d * H_ + t * 16 + col];
    w2v[t] = w2cat[d * H_ + t * 16 + col];
  }
  float b2v = b2cat[d];

#pragma unroll
  for (int g = 0; g < 2; ++g) {
#pragma unroll
    for (int r = 0; r < 8; ++r) {
      float s = 0.0f;
#pragma unroll
      for (int t = 0; t < 5; ++t)
        s += fmaxf(acc[g][t][r] + b1v[t], 0.0f) * w2v[t];
      // reduce over the 16 lanes of each half (they cover n = 0..79 exactly)
      s += __shfl_xor(s, 1, 32);
      s += __shfl_xor(s, 2, 32);
      s += __shfl_xor(s, 4, 32);
      s += __shfl_xor(s, 8, 32);
      if (col == 0) {
        int m = g * 16 + r + 8 * half;
        int b = sidx[m];
        if (b >= 0) out[b] += s + b2v;   // out already holds domain_logits
      }
    }
  }
}

// ---------------------------------------------------------------------------
extern "C" void kernel_launch(void* const* d_in, const int* in_sizes, int n_in,
                              void* d_out, int out_size, void* d_ws, size_t ws_size,
                              hipStream_t stream) {
  (void)in_sizes; (void)n_in; (void)out_size; (void)ws_size;

  const int*   pid       = (const int*)  d_in[0];
  const int*   feats     = (const int*)  d_in[1];
  const float* emb_pid   = (const float*)d_in[2];
  const float* emb_feats = (const float*)d_in[3];
  const float* sw1       = (const float*)d_in[4];
  const float* dw1       = (const float*)d_in[5];
  const float* sb1       = (const float*)d_in[6];
  const float* db1       = (const float*)d_in[7];
  const float* sw2       = (const float*)d_in[8];
  const float* dw2       = (const float*)d_in[9];
  const float* sb2       = (const float*)d_in[10];
  const float* db2       = (const float*)d_in[11];
  const float* dlw       = (const float*)d_in[12];
  const float* dlb       = (const float*)d_in[13];
  float* out = (float*)d_out;

  // Workspace layout (~2.5 MB)
  int*    cnt   = (int*)d_ws;                  // 16 ints (8 used)
  int*    perm  = cnt + 16;                    // D*B ints
  float2* w1p   = (float2*)(perm + D_ * B_);   // D*KP*H float2
  float*  b1cat = (float*)(w1p + D_ * KP_ * H_); // D*H
  float*  w2cat = b1cat + D_ * H_;             // D*H
  float*  b2cat = w2cat + D_ * H_;             // D

  starnet_prep<<<(D_ * KP_ * H_ + 255) / 256, 256, 0, stream>>>(
      sw1, dw1, sb1, db1, sw2, dw2, sb2, db2, w1p, b1cat, w2cat, b2cat, cnt);

  starnet_route<<<B_ / 256, 256, 0, stream>>>(
      pid, emb_pid, dlw, dlb, out, cnt, perm);

  starnet_gemm<<<dim3(B_ / TM_, D_), 32, 0, stream>>>(
      pid, feats, emb_pid, emb_feats, w1p, b1cat, w2cat, b2cat, cnt, perm, out);
}